// DPOTNet3D_HydraLora_14259291423368
// MI455X (gfx1250) — compile-verified
//
#include <hip/hip_runtime.h>
#include <hip/hip_bf16.h>

// ---------------------------------------------------------------------------
// DPOTNet3D HydraLoRA forward for MI455X (gfx1250, wave32).
//
// Strategy:
//  * Truncated 3D rFFT == dense DFT GEMMs per axis (only 32/32/8 modes kept).
//  * All GEMMs run on the fp32 matrix pipe via V_WMMA_F32_16X16X4_F32.
//  * HydraLoRA collapses exactly:  W_eff = w + (1/R) * A @ (sum_e ew[e]*B[e]),
//    so the spectral mixer is a fused complex block-diag matmul+GELU+matmul.
//  * Traffic ~1 GB total -> ~66us @ 23.3 TB/s; ~19 GFLOP fp32 on WMMA pipe.
//  * Workspace layout (floats), total ~67.15M floats (~269 MB of d_ws):
//      Y1(r,i) 2*16.78M  (reused as Z1 for inverse)   [B,H,W,8,C]
//      Y2(r,i) 2* 8.39M  (reused as Z2)               [B,H,32,8,C]
//      S (r,i) 2* 4.19M                                [B,32,32,8,C]
//      O (r,i) 2* 4.19M                                [B,32,32,8,C]
//      Weff    4*8192, trig fragment tables 9216
// ---------------------------------------------------------------------------

typedef __attribute__((ext_vector_type(2))) float v2f;
typedef __attribute__((ext_vector_type(8))) float v8f;

__device__ __forceinline__ v8f wmma_f32(v2f a, v2f b, v8f c) {
  // D(16x16,f32) = A(16x4,f32) * B(4x16,f32) + C
  return __builtin_amdgcn_wmma_f32_16x16x4_f32(
      false, a, false, b, (short)0, c, false, false);
}

__device__ __forceinline__ v2f vneg(v2f a) { v2f r; r.x = -a.x; r.y = -a.y; return r; }

__device__ __forceinline__ float gelu_exact(float x) {
  return 0.5f * x * (1.0f + erff(x * 0.70710678118654752440f));
}

// ---------------- constants ----------------
#define Bb   2
#define Hh   64
#define Wd   64
#define Ll   32
#define Cc   256
#define TM   8      // kept rfft modes along L
#define NBk  8
#define BSz  32

// trig table sub-offsets (in floats, each entry is a float2 per lane)
#define OFF_TL   0          // [8 steps][32 lanes][2]           forward L (incl fwd norm)
#define OFF_TC   512        // [2 mt][16 steps][32][2]          cos(2pi kw/64)
#define OFF_TS   2560       // [2 mt][16 steps][32][2]          sin(2pi kw/64)
#define OFF_TIC  4608       // [4 mt][8 steps][32][2]           cos(2pi hk/64)
#define OFF_TIS  6656       // [4 mt][8 steps][32][2]           sin(2pi hk/64)
#define OFF_TIL  8704       // [2 mt][4 steps][32][2]           irfft L (incl inv norm)

// ---------------------------------------------------------------------------
// Trig fragment tables. A-fragment layout (ISA 7.12.2, f32 16x16x4):
//   lane L holds rows r = L&15;  a.x = A[r][4s + 2*(L>=16)], a.y = A[r][..+1]
// ---------------------------------------------------------------------------
__global__ void k_init_tables(float* __restrict__ tab) {
  const int   tid    = threadIdx.x;
  const float TWO_PI = 6.28318530717958647692f;
  const float FN     = 0.00276213586400995107f;  // 1/sqrt(64*64*32)

  // tL: rows 0..7 = cos(2pi r l/32)*FN ; rows 8..15 = -sin(2pi (r-8) l/32)*FN
  for (int i = tid; i < 8 * 32; i += blockDim.x) {
    int s = i >> 5, lane = i & 31;
    int row = lane & 15, khalf = (lane >> 4) << 1;
    for (int j = 0; j < 2; ++j) {
      int l = 4 * s + khalf + j;
      float v;
      if (row < 8) v =  cosf(TWO_PI * (float)((row * l) & 31) * (1.0f / 32.0f)) * FN;
      else         v = -sinf(TWO_PI * (float)(((row - 8) * l) & 31) * (1.0f / 32.0f)) * FN;
      tab[OFF_TL + i * 2 + j] = v;
    }
  }
  // tC/tS: A[k][w], k = mt*16+(lane&15) in 0..31, w = 4s+khalf+j in 0..63
  for (int i = tid; i < 2 * 16 * 32; i += blockDim.x) {
    int mt = i >> 9, s = (i >> 5) & 15, lane = i & 31;
    int k = mt * 16 + (lane & 15), khalf = (lane >> 4) << 1;
    for (int j = 0; j < 2; ++j) {
      int w = 4 * s + khalf + j;
      float ang = TWO_PI * (float)((k * w) & 63) * (1.0f / 64.0f);
      tab[OFF_TC + i * 2 + j] = cosf(ang);
      tab[OFF_TS + i * 2 + j] = sinf(ang);
    }
  }
  // tiC/tiS: A[h][k], h = mt*16+(lane&15) in 0..63, k = 4s+khalf+j in 0..31
  for (int i = tid; i < 4 * 8 * 32; i += blockDim.x) {
    int mt = i >> 8, s = (i >> 5) & 7, lane = i & 31;
    int h = mt * 16 + (lane & 15), khalf = (lane >> 4) << 1;
    for (int j = 0; j < 2; ++j) {
      int kk = 4 * s + khalf + j;
      float ang = TWO_PI * (float)((h * kk) & 63) * (1.0f / 64.0f);
      tab[OFF_TIC + i * 2 + j] = cosf(ang);
      tab[OFF_TIS + i * 2 + j] = sinf(ang);
    }
  }
  // tiL: A[l][row16]; rows 0..7 act on Xr with wk*cos; rows 8..15 on Xi with -wk*sin
  for (int i = tid; i < 2 * 4 * 32; i += blockDim.x) {
    int mt = i >> 7, s = (i >> 5) & 3, lane = i & 31;
    int l = mt * 16 + (lane & 15), khalf = (lane >> 4) << 1;
    for (int j = 0; j < 2; ++j) {
      int row = 4 * s + khalf + j;
      float v;
      if (row < 8) {
        int k = row; float wk = (k == 0) ? 1.0f : 2.0f;
        v =  wk * cosf(TWO_PI * (float)((k * l) & 31) * (1.0f / 32.0f)) * FN;
      } else {
        int k = row - 8; float wk = (k == 0) ? 1.0f : 2.0f;
        v = -wk * sinf(TWO_PI * (float)((k * l) & 31) * (1.0f / 32.0f)) * FN;
      }
      tab[OFF_TIL + i * 2 + j] = v;
    }
  }
}

// ---------------------------------------------------------------------------
// Effective weights: Wout[set][n][c][o] = w[n,c,o] + (1/R) sum_r A[n,c,r]*Beff[r,o]
// sets: 0:(w1[0],a1r,B1r) 1:(w1[1],a1i,B1i) 2:(w2[0],a2r,B2r) 3:(w2[1],a2i,B2i)
// ---------------------------------------------------------------------------
__global__ void k_weights(const float* __restrict__ w1, const float* __restrict__ w2,
                          const float* __restrict__ a1r, const float* __restrict__ B1r,
                          const float* __restrict__ a1i, const float* __restrict__ B1i,
                          const float* __restrict__ a2r, const float* __restrict__ B2r,
                          const float* __restrict__ a2i, const float* __restrict__ B2i,
                          const float* __restrict__ ew, float* __restrict__ wout) {
  const int set = blockIdx.y, n = blockIdx.x;
  const int o = threadIdx.x, c = threadIdx.y;
  const float *w, *A, *Bm;
  switch (set) {
    case 0:  w = w1;        A = a1r; Bm = B1r; break;
    case 1:  w = w1 + 8192; A = a1i; Bm = B1i; break;
    case 2:  w = w2;        A = a2r; Bm = B2r; break;
    default: w = w2 + 8192; A = a2i; Bm = B2i; break;
  }
  const float e0 = ew[0], e1 = ew[1], e2 = ew[2], e3 = ew[3];
  float acc = 0.0f;
  for (int r = 0; r < 32; ++r) {
    const float* bp = Bm + (long)n * 4096 + r * 32 + o;  // [n][e][r][o], e-stride 1024
    float beff = e0 * bp[0] + e1 * bp[1024] + e2 * bp[2048] + e3 * bp[3072];
    acc += A[(n * 32 + c) * 32 + r] * beff;
  }
  wout[set * 8192 + n * 1024 + c * 32 + o] =
      w[n * 1024 + c * 32 + o] + acc * (1.0f / 32.0f);
}

// ---------------------------------------------------------------------------
// Stage 1: truncated rDFT along L. Per wave: D[16(8r+8i) x 16ch] = T[16x32]@x[32x16]
// ---------------------------------------------------------------------------
__global__ void k_fwd_L(const float* __restrict__ x, const float* __restrict__ tL,
                        float* __restrict__ y1r, float* __restrict__ y1i) {
  const int lane  = threadIdx.x & 31;
  const int wid   = blockIdx.x * 8 + (threadIdx.x >> 5);
  const int ct    = wid & 15;
  const long pos  = wid >> 4;                 // (b*H+h)*W + w, 0..8191
  const int col   = lane & 15;
  const int khalf = (lane >> 4) << 1;
  const v2f* t2   = (const v2f*)tL;
  const long ib   = pos * (Ll * Cc) + ct * 16 + col;
  v8f d = {};
#pragma unroll
  for (int s = 0; s < 8; ++s) {
    v2f a = t2[s * 32 + lane];
    const int k0 = 4 * s + khalf;
    v2f bf;
    bf.x = x[ib + (long)k0 * Cc];
    bf.y = x[ib + (long)(k0 + 1) * Cc];
    d = wmma_f32(a, bf, d);
  }
  // rows 0..7 -> Y1r[k], rows 8..15 -> Y1i[k]
  float* outp = (lane < 16) ? y1r : y1i;
  const long ob = pos * (TM * Cc) + ct * 16 + col;
#pragma unroll
  for (int v = 0; v < 8; ++v) outp[ob + (long)v * Cc] = d[v];
}

// ---------------------------------------------------------------------------
// Stage 2: complex DFT along W (64 -> 32 modes). X_r = C@xr + S@xi ; X_i = C@xi - S@xr
// ---------------------------------------------------------------------------
__global__ void k_fwd_W(const float* __restrict__ inr, const float* __restrict__ ini,
                        const float* __restrict__ tC, const float* __restrict__ tS,
                        float* __restrict__ outr, float* __restrict__ outi) {
  const int lane = threadIdx.x & 31;
  const int wid  = blockIdx.x * 8 + (threadIdx.x >> 5);
  const int ct = wid & 15, mt = (wid >> 4) & 1;
  const int r  = wid >> 5;             // (b*H+h)*8 + kl
  const int kl = r & 7;  const long bh = r >> 3;
  const int col = lane & 15, khalf = (lane >> 4) << 1, hi8 = (lane >> 4) << 3;
  const long ib = bh * 131072 + kl * 256 + ct * 16 + col;  // Y1 [bh][w][kl][c]
  const long ob = bh * 65536  + kl * 256 + ct * 16 + col;  // Y2 [bh][kw][kl][c]
  const v2f* tc2 = (const v2f*)tC; const v2f* ts2 = (const v2f*)tS;
  v8f dr = {}, di = {};
#pragma unroll
  for (int s = 0; s < 16; ++s) {
    const int ti = (mt * 16 + s) * 32 + lane;
    v2f ac = tc2[ti], as = ts2[ti];
    const int k0 = 4 * s + khalf;
    v2f br, bi;
    br.x = inr[ib + (long)k0 * 2048]; br.y = inr[ib + (long)(k0 + 1) * 2048];
    bi.x = ini[ib + (long)k0 * 2048]; bi.y = ini[ib + (long)(k0 + 1) * 2048];
    dr = wmma_f32(ac, br, dr);
    dr = wmma_f32(as, bi, dr);
    di = wmma_f32(ac, bi, di);
    di = wmma_f32(vneg(as), br, di);
  }
#pragma unroll
  for (int v = 0; v < 8; ++v) {
    const int m = mt * 16 + v + hi8;
    outr[ob + (long)m * 2048] = dr[v];
    outi[ob + (long)m * 2048] = di[v];
  }
}

// Stage 3: complex DFT along H (64 -> 32 modes), same math, different strides.
__global__ void k_fwd_H(const float* __restrict__ inr, const float* __restrict__ ini,
                        const float* __restrict__ tC, const float* __restrict__ tS,
                        float* __restrict__ outr, float* __restrict__ outi) {
  const int lane = threadIdx.x & 31;
  const int wid  = blockIdx.x * 8 + (threadIdx.x >> 5);
  const int ct = wid & 15, mt = (wid >> 4) & 1;
  const int r  = wid >> 5;                 // ((b*32+kw)*8 + kl)
  const int kl = r & 7; const int t = r >> 3;
  const int kw = t & 31; const long b = t >> 5;
  const int col = lane & 15, khalf = (lane >> 4) << 1, hi8 = (lane >> 4) << 3;
  const long ib = b * 4194304 + kw * 2048 + kl * 256 + ct * 16 + col;  // Y2 [b][h][kw][kl][c]
  const long ob = b * 2097152 + kw * 2048 + kl * 256 + ct * 16 + col;  // S  [b][kh][kw][kl][c]
  const v2f* tc2 = (const v2f*)tC; const v2f* ts2 = (const v2f*)tS;
  v8f dr = {}, di = {};
#pragma unroll
  for (int s = 0; s < 16; ++s) {
    const int ti = (mt * 16 + s) * 32 + lane;
    v2f ac = tc2[ti], as = ts2[ti];
    const int k0 = 4 * s + khalf;
    v2f br, bi;
    br.x = inr[ib + (long)k0 * 65536]; br.y = inr[ib + (long)(k0 + 1) * 65536];
    bi.x = ini[ib + (long)k0 * 65536]; bi.y = ini[ib + (long)(k0 + 1) * 65536];
    dr = wmma_f32(ac, br, dr);
    dr = wmma_f32(as, bi, dr);
    di = wmma_f32(ac, bi, di);
    di = wmma_f32(vneg(as), br, di);
  }
#pragma unroll
  for (int v = 0; v < 8; ++v) {
    const int m = mt * 16 + v + hi8;
    outr[ob + (long)m * 65536] = dr[v];
    outi[ob + (long)m * 65536] = di[v];
  }
}

// ---------------------------------------------------------------------------
// Fused spectral mixer: per wave = one channel-block n, 16 spectral positions.
//   z = s @ (W1r + i W1i) + (b1r + i b1i)  ; gelu on re/im ; repeat with layer 2
// LDS used only for D-fragment -> A-fragment transpose between the layers.
// ---------------------------------------------------------------------------
__global__ void k_spec(const float* __restrict__ sr, const float* __restrict__ si,
                       float* __restrict__ outr, float* __restrict__ outi,
                       const float* __restrict__ wbuf,
                       const float* __restrict__ b1, const float* __restrict__ b2) {
  __shared__ float ldsr[8][16][32];
  __shared__ float ldsi[8][16][32];
  const int lane = threadIdx.x & 31;
  const int n    = threadIdx.x >> 5;        // channel block 0..7
  const int p0   = blockIdx.x * 16;         // spectral position tile
  const int row  = lane & 15;
  const int khalf = (lane >> 4) << 1, hi8 = (lane >> 4) << 3;
  const float* W1r = wbuf + 0 * 8192 + n * 1024;
  const float* W1i = wbuf + 1 * 8192 + n * 1024;
  const float* W2r = wbuf + 2 * 8192 + n * 1024;
  const float* W2i = wbuf + 3 * 8192 + n * 1024;

  // ---- layer 1 ----
  for (int nt = 0; nt < 2; ++nt) {
    v8f zr = {}, zi = {};
    const int colg = nt * 16 + row;
#pragma unroll
    for (int s = 0; s < 8; ++s) {
      const int k0 = 4 * s + khalf;
      const long ab = (long)(p0 + row) * 256 + n * 32 + k0;
      v2f ar = *(const v2f*)(sr + ab);
      v2f ai = *(const v2f*)(si + ab);
      v2f bwr, bwi;
      bwr.x = W1r[k0 * 32 + colg]; bwr.y = W1r[(k0 + 1) * 32 + colg];
      bwi.x = W1i[k0 * 32 + colg]; bwi.y = W1i[(k0 + 1) * 32 + colg];
      zr = wmma_f32(ar, bwr, zr);
      zr = wmma_f32(vneg(ai), bwi, zr);
      zi = wmma_f32(ai, bwr, zi);
      zi = wmma_f32(ar, bwi, zi);
    }
    const float br1 = b1[n * 32 + colg];
    const float bi1 = b1[256 + n * 32 + colg];
    const float cr = br1 - bi1, ci = br1 + bi1;
#pragma unroll
    for (int v = 0; v < 8; ++v) {
      ldsr[n][v + hi8][colg] = gelu_exact(zr[v] + cr);
      ldsi[n][v + hi8][colg] = gelu_exact(zi[v] + ci);
    }
  }
  // same-wave LDS ops are in-order; each wave touches only its own slice.

  // ---- layer 2 ----
  for (int nt = 0; nt < 2; ++nt) {
    v8f zr = {}, zi = {};
    const int colg = nt * 16 + row;
#pragma unroll
    for (int s = 0; s < 8; ++s) {
      const int k0 = 4 * s + khalf;
      v2f ar, ai, bwr, bwi;
      ar.x = ldsr[n][row][k0]; ar.y = ldsr[n][row][k0 + 1];
      ai.x = ldsi[n][row][k0]; ai.y = ldsi[n][row][k0 + 1];
      bwr.x = W2r[k0 * 32 + colg]; bwr.y = W2r[(k0 + 1) * 32 + colg];
      bwi.x = W2i[k0 * 32 + colg]; bwi.y = W2i[(k0 + 1) * 32 + colg];
      zr = wmma_f32(ar, bwr, zr);
      zr = wmma_f32(vneg(ai), bwi, zr);
      zi = wmma_f32(ai, bwr, zi);
      zi = wmma_f32(ar, bwi, zi);
    }
    const float br2 = b2[n * 32 + colg];
    const float bi2 = b2[256 + n * 32 + colg];
    const float cr = br2 - bi2, ci = br2 + bi2;
#pragma unroll
    for (int v = 0; v < 8; ++v) {
      const long oidx = (long)(p0 + v + hi8) * 256 + n * 32 + colg;
      outr[oidx] = zr[v] + cr;
      outi[oidx] = zi[v] + ci;
    }
  }
}

// ---------------------------------------------------------------------------
// Inverse H (32 modes -> 64). x_r = C@Xr - S@Xi ; x_i = C@Xi + S@Xr
// ---------------------------------------------------------------------------
__global__ void k_inv_H(const float* __restrict__ inr, const float* __restrict__ ini,
                        const float* __restrict__ tiC, const float* __restrict__ tiS,
                        float* __restrict__ outr, float* __restrict__ outi) {
  const int lane = threadIdx.x & 31;
  const int wid  = blockIdx.x * 8 + (threadIdx.x >> 5);
  const int ct = wid & 15, mt = (wid >> 4) & 3;
  const int r  = wid >> 6;                 // ((b*32+kw)*8 + kl)
  const int kl = r & 7; const int t = r >> 3;
  const int kw = t & 31; const long b = t >> 5;
  const int col = lane & 15, khalf = (lane >> 4) << 1, hi8 = (lane >> 4) << 3;
  const long ib = b * 2097152 + kw * 2048 + kl * 256 + ct * 16 + col;  // O  [b][kh][kw][kl][c]
  const long ob = b * 4194304 + kw * 2048 + kl * 256 + ct * 16 + col;  // Z2 [b][h][kw][kl][c]
  const v2f* tc2 = (const v2f*)tiC; const v2f* ts2 = (const v2f*)tiS;
  v8f dr = {}, di = {};
#pragma unroll
  for (int s = 0; s < 8; ++s) {
    const int ti = (mt * 8 + s) * 32 + lane;
    v2f ac = tc2[ti], as = ts2[ti];
    const int k0 = 4 * s + khalf;
    v2f br, bi;
    br.x = inr[ib + (long)k0 * 65536]; br.y = inr[ib + (long)(k0 + 1) * 65536];
    bi.x = ini[ib + (long)k0 * 65536]; bi.y = ini[ib + (long)(k0 + 1) * 65536];
    dr = wmma_f32(ac, br, dr);
    dr = wmma_f32(vneg(as), bi, dr);
    di = wmma_f32(ac, bi, di);
    di = wmma_f32(as, br, di);
  }
#pragma unroll
  for (int v = 0; v < 8; ++v) {
    const int m = mt * 16 + v + hi8;
    outr[ob + (long)m * 65536] = dr[v];
    outi[ob + (long)m * 65536] = di[v];
  }
}

// Inverse W (32 modes -> 64).
__global__ void k_inv_W(const float* __restrict__ inr, const float* __restrict__ ini,
                        const float* __restrict__ tiC, const float* __restrict__ tiS,
                        float* __restrict__ outr, float* __restrict__ outi) {
  const int lane = threadIdx.x & 31;
  const int wid  = blockIdx.x * 8 + (threadIdx.x >> 5);
  const int ct = wid & 15, mt = (wid >> 4) & 3;
  const int r  = wid >> 6;                 // (b*H+h)*8 + kl
  const int kl = r & 7; const long bh = r >> 3;
  const int col = lane & 15, khalf = (lane >> 4) << 1, hi8 = (lane >> 4) << 3;
  const long ib = bh * 65536  + kl * 256 + ct * 16 + col;  // Z2 [bh][kw][kl][c]
  const long ob = bh * 131072 + kl * 256 + ct * 16 + col;  // Z1 [bh][w][kl][c]
  const v2f* tc2 = (const v2f*)tiC; const v2f* ts2 = (const v2f*)tiS;
  v8f dr = {}, di = {};
#pragma unroll
  for (int s = 0; s < 8; ++s) {
    const int ti = (mt * 8 + s) * 32 + lane;
    v2f ac = tc2[ti], as = ts2[ti];
    const int k0 = 4 * s + khalf;
    v2f br, bi;
    br.x = inr[ib + (long)k0 * 2048]; br.y = inr[ib + (long)(k0 + 1) * 2048];
    bi.x = ini[ib + (long)k0 * 2048]; bi.y = ini[ib + (long)(k0 + 1) * 2048];
    dr = wmma_f32(ac, br, dr);
    dr = wmma_f32(vneg(as), bi, dr);
    di = wmma_f32(ac, bi, di);
    di = wmma_f32(as, br, di);
  }
#pragma unroll
  for (int v = 0; v < 8; ++v) {
    const int m = mt * 16 + v + hi8;
    outr[ob + (long)m * 2048] = dr[v];
    outi[ob + (long)m * 2048] = di[v];
  }
}

// Inverse irfft along L (8 complex modes -> 32 reals) + residual add.
// B-matrix rows 0..7 = Z1r[k], rows 8..15 = Z1i[k]; coefficients in tiL.
__global__ void k_inv_L(const float* __restrict__ z1r, const float* __restrict__ z1i,
                        const float* __restrict__ x, const float* __restrict__ tiL,
                        float* __restrict__ out) {
  const int lane = threadIdx.x & 31;
  const int wid  = blockIdx.x * 8 + (threadIdx.x >> 5);
  const int ct = wid & 15, mt = (wid >> 4) & 1;
  const long pos = wid >> 5;               // (b*H+h)*W + w
  const int col = lane & 15, khalf = (lane >> 4) << 1, hi8 = (lane >> 4) << 3;
  const v2f* t2 = (const v2f*)tiL;
  const long ib = pos * 2048 + ct * 16 + col;
  v8f d = {};
#pragma unroll
  for (int s = 0; s < 4; ++s) {
    v2f a = t2[(mt * 4 + s) * 32 + lane];
    const int r0 = 4 * s + khalf;          // even, so (r0,r0+1) stay in same half
    v2f bf;
    bf.x = (r0 < 8) ? z1r[ib + (long)r0 * 256] : z1i[ib + (long)(r0 - 8) * 256];
    bf.y = (r0 < 8) ? z1r[ib + (long)(r0 + 1) * 256] : z1i[ib + (long)(r0 - 7) * 256];
    d = wmma_f32(a, bf, d);
  }
  const long ob = pos * 8192 + ct * 16 + col;
#pragma unroll
  for (int v = 0; v < 8; ++v) {
    const int m = mt * 16 + v + hi8;
    out[ob + (long)m * 256] = d[v] + x[ob + (long)m * 256];
  }
}

// ---------------------------------------------------------------------------
extern "C" void kernel_launch(void* const* d_in, const int* in_sizes, int n_in,
                              void* d_out, int out_size, void* d_ws, size_t ws_size,
                              hipStream_t stream) {
  (void)in_sizes; (void)n_in; (void)out_size; (void)ws_size;

  const float* x   = (const float*)d_in[0];
  const float* w1  = (const float*)d_in[1];
  const float* b1  = (const float*)d_in[2];
  const float* w2  = (const float*)d_in[3];
  const float* b2  = (const float*)d_in[4];
  const float* a1r = (const float*)d_in[5];
  const float* B1r = (const float*)d_in[6];
  const float* a1i = (const float*)d_in[7];
  const float* B1i = (const float*)d_in[8];
  const float* a2r = (const float*)d_in[9];
  const float* B2r = (const float*)d_in[10];
  const float* a2i = (const float*)d_in[11];
  const float* B2i = (const float*)d_in[12];
  const float* ew  = (const float*)d_in[13];
  float* out = (float*)d_out;

  float* ws = (float*)d_ws;
  const long N1 = 16777216;  // B*H*W*TM*C
  const long N2 = 8388608;   // B*H*32*TM*C
  const long N3 = 4194304;   // B*32*32*TM*C
  float* y1r = ws;                 float* y1i = ws + N1;
  float* y2r = ws + 2 * N1;        float* y2i = y2r + N2;
  float* sr  = ws + 2 * N1 + 2 * N2;  float* si = sr + N3;
  float* orr = si + N3;            float* oi = orr + N3;
  float* wbuf = oi + N3;           // 4*8192 effective weights
  float* tab  = wbuf + 32768;      // 9216 floats of trig fragments
  float* z2r = y2r; float* z2i = y2i;   // reuse (Y2 dead after k_fwd_H)
  float* z1r = y1r; float* z1i = y1i;   // reuse (Y1 dead after k_fwd_W)

  float* tL  = tab + OFF_TL;
  float* tC  = tab + OFF_TC;
  float* tS  = tab + OFF_TS;
  float* tiC = tab + OFF_TIC;
  float* tiS = tab + OFF_TIS;
  float* tiL = tab + OFF_TIL;

  k_init_tables<<<1, 256, 0, stream>>>(tab);
  k_weights<<<dim3(NBk, 4), dim3(32, 32), 0, stream>>>(
      w1, w2, a1r, B1r, a1i, B1i, a2r, B2r, a2i, B2i, ew, wbuf);

  // forward truncated DFT
  k_fwd_L<<<16384, 256, 0, stream>>>(x, tL, y1r, y1i);        // 131072 waves
  k_fwd_W<<<4096, 256, 0, stream>>>(y1r, y1i, tC, tS, y2r, y2i);
  k_fwd_H<<<2048, 256, 0, stream>>>(y2r, y2i, tC, tS, sr, si);

  // fused HydraLoRA complex mixer in truncated-mode domain (L2 resident)
  k_spec<<<1024, 256, 0, stream>>>(sr, si, orr, oi, wbuf, b1, b2);

  // inverse transform + residual
  k_inv_H<<<4096, 256, 0, stream>>>(orr, oi, tiC, tiS, z2r, z2i);
  k_inv_W<<<8192, 256, 0, stream>>>(z2r, z2i, tiC, tiS, z1r, z1i);
  k_inv_L<<<32768, 256, 0, stream>>>(z1r, z1i, x, tiL, out);
}